// EfficientMultiTensorProductGauntS2Grid_58609123721877
// MI455X (gfx1250) — compile-verified
//
#include <hip/hip_runtime.h>
#include <hip/hip_bf16.h>

#define PI_D 3.14159265358979323846

typedef __attribute__((ext_vector_type(2))) float v2f;
typedef __attribute__((ext_vector_type(8))) float v8f;

// D = A(16x4 f32) * B(4x16 f32) + C(16x16 f32), wave32 WMMA
__device__ __forceinline__ v8f wmma_f32(v2f a, v2f b, v8f c) {
    return __builtin_amdgcn_wmma_f32_16x16x4_f32(
        /*neg_a=*/false, a, /*neg_b=*/false, b,
        /*c_mod=*/(short)0, c, /*reuse_a=*/false, /*reuse_b=*/false);
}

__device__ __forceinline__ v2f ld2(const float* p) {
    return *(const v2f*)p;   // 8B-aligned by construction (even float offsets)
}

// ---------------------------------------------------------------------------
// Setup: build S2 grid matrices Y (36x144) and Yw (144x48, zero-padded) and
// store them in WMMA-B-fragment-packed layout in workspace:
//   Ypk [nt(9)][ks(9)][lane(32)][2]   for to_grid   (K = SH dim, N = grid pt)
//   Ywpk[nt(3)][ks(36)][lane(32)][2]  for from_grid (K = grid pt, N = SH dim)
// Fragment convention (f32 16x16x4): VGPR0 holds K=2*(lane/16), VGPR1 = K+1.
// ---------------------------------------------------------------------------
__global__ void gaunt_setup(float* __restrict__ Ypk, float* __restrict__ Ywpk) {
    __shared__ float Yd[36 * 144];   // [k][p]
    __shared__ float Ywd[144 * 48];  // [p][k'] padded
    const int tid = threadIdx.x;     // 256 threads

    if (tid < 144) {
        const int b = tid / 12, a = tid % 12;
        const int n = 12;
        // Gauss-Legendre node b via Newton on P_12
        double x = cos(PI_D * (b + 0.75) / (n + 0.5));
        double p0, p1, dp;
        for (int it = 0; it < 60; ++it) {
            p0 = 1.0; p1 = x;
            for (int k = 2; k <= n; ++k) {
                double pk = ((2.0 * k - 1.0) * x * p1 - (k - 1.0) * p0) / k;
                p0 = p1; p1 = pk;
            }
            dp = n * (x * p1 - p0) / (x * x - 1.0);
            double dx = p1 / dp;
            x -= dx;
            if (fabs(dx) < 1e-14) break;
        }
        p0 = 1.0; p1 = x;
        for (int k = 2; k <= n; ++k) {
            double pk = ((2.0 * k - 1.0) * x * p1 - (k - 1.0) * p0) / k;
            p0 = p1; p1 = pk;
        }
        dp = n * (x * p1 - p0) / (x * x - 1.0);
        const double wq = 2.0 / ((1.0 - x * x) * dp * dp);

        // fully-normalized associated Legendre Pbar_l^m(x)
        double P[6][6];
        const double s = sqrt(fmax(1.0 - x * x, 0.0));
        P[0][0] = sqrt(1.0 / (4.0 * PI_D));
        for (int m = 1; m <= 5; ++m)
            P[m][m] = sqrt((2.0 * m + 1.0) / (2.0 * m)) * s * P[m - 1][m - 1];
        for (int m = 0; m <= 4; ++m)
            P[m + 1][m] = sqrt(2.0 * m + 3.0) * x * P[m][m];
        for (int m = 0; m <= 5; ++m)
            for (int l = m + 2; l <= 5; ++l) {
                double aa = sqrt((4.0 * l * l - 1.0) / ((double)l * l - (double)m * m));
                double bb = sqrt(((l - 1.0) * (l - 1.0) - (double)m * m) /
                                 (4.0 * (l - 1.0) * (l - 1.0) - 1.0));
                P[l][m] = aa * (x * P[l - 1][m] - bb * P[l - 2][m]);
            }

        const double alpha = a * (2.0 * PI_D / 12.0);
        const double sqrt2 = sqrt(2.0);
        const int p = b * 12 + a;
        const double wfac = wq * (2.0 * PI_D / 12.0);
        for (int j = 36; j < 48; ++j) Ywd[p * 48 + j] = 0.0f;  // pad cols
        for (int l = 0; l <= 5; ++l)
            for (int m = -l; m <= l; ++m) {
                const int idx = l * l + l + m;
                double ang;
                if (m > 0)      ang = sqrt2 * cos((double)m * alpha);
                else if (m == 0) ang = 1.0;
                else             ang = sqrt2 * sin((double)(-m) * alpha);
                const double y = P[l][m < 0 ? -m : m] * ang;
                Yd[idx * 144 + p] = (float)y;
                Ywd[p * 48 + idx] = (float)(y * wfac);
            }
    }
    __syncthreads();

    // pack to B-fragment layout
    for (int i = tid; i < 9 * 9 * 32; i += 256) {
        const int lane = i & 31, t = i >> 5;
        const int ks = t % 9, nt = t / 9;
        const int half = lane >> 4, lm = lane & 15;
        const int k0 = ks * 4 + 2 * half, col = nt * 16 + lm;
        Ypk[i * 2 + 0] = Yd[k0 * 144 + col];
        Ypk[i * 2 + 1] = Yd[(k0 + 1) * 144 + col];
    }
    for (int i = tid; i < 3 * 36 * 32; i += 256) {
        const int lane = i & 31, t = i >> 5;
        const int ks = t % 36, nt = t / 36;
        const int half = lane >> 4, lm = lane & 15;
        const int k0 = ks * 4 + 2 * half, col = nt * 16 + lm;
        Ywpk[i * 2 + 0] = Ywd[k0 * 48 + col];
        Ywpk[i * 2 + 1] = Ywd[(k0 + 1) * 48 + col];
    }
}

// ---------------------------------------------------------------------------
// Main fused kernel: one block = (atom, 64-channel half); 128 threads = 4
// waves; wave w owns channel M-tile [w*16, w*16+16). Everything stays in LDS.
// ---------------------------------------------------------------------------
__global__ __launch_bounds__(128) void gaunt_main(
        const float* __restrict__ feat,    // [4096][128][36]
        const float* __restrict__ atype,   // [4096][10]
        const float* __restrict__ w,       // [3][10][128][6]
        const float* __restrict__ Ypk,     // packed 5184
        const float* __restrict__ Ywpk,    // packed 6912
        float* __restrict__ out) {         // [4096][128][36]
    const int tid  = threadIdx.x;
    const int lane = tid & 31;
    const int wave = tid >> 5;       // 0..3
    const int half = lane >> 4;      // fragment half
    const int lm   = lane & 15;
    const int atom = (int)blockIdx.x >> 1;
    const int ch0  = ((int)blockIdx.x & 1) * 64;
    const int wrow = wave * 16;      // local channel row base for this wave

    extern __shared__ float smem[];
    float* sYpk  = smem;              // 5184
    float* sYwpk = sYpk + 5184;       // 6912
    float* sP    = sYwpk + 6912;      // 64*36  = 2304  (prod coefficients)
    float* sR    = sP + 2304;         // 2304           (result accumulator)
    float* sW    = sR + 2304;         // 3*64*6 = 1152  (atom-weighted w)
    float* sAT   = sW + 1152;         // 16             (atom_type row)
    float* sFG   = sAT + 16;          // 64*148 = 9472  (feat grid, stride 148)
    float* sPG   = sFG + 9472;        // 9472           (prod grid)
    // total 36816 floats = 147264 bytes

    const size_t fbase = ((size_t)atom * 128 + ch0) * 36;
    __builtin_prefetch(feat + fbase, 0, 0);  // global_prefetch_b8

    // ---- stage 0: fill LDS -------------------------------------------------
    for (int i = tid; i < 5184; i += 128) sYpk[i] = Ypk[i];
    for (int i = tid; i < 6912; i += 128) sYwpk[i] = Ywpk[i];
    for (int i = tid; i < 2304; i += 128) sP[i] = feat[fbase + i];
    if (tid < 10) sAT[tid] = atype[atom * 10 + tid];
    __syncthreads();

    // ---- atom-type-weighted per-l weights: sW[nu][c][l] --------------------
    for (int i = tid; i < 3 * 64 * 6; i += 128) {
        const int nu = i / 384, rem = i % 384, c = rem / 6, l = rem % 6;
        float acc = 0.0f;
        #pragma unroll
        for (int e = 0; e < 10; ++e)
            acc += sAT[e] * w[(((nu * 10 + e) * 128) + ch0 + c) * 6 + l];
        sW[i] = acc;
    }
    __syncthreads();

    // ---- stage A: feat_grid = to_grid(atom_feat) ---------------------------
    for (int nt = 0; nt < 9; ++nt) {
        v8f acc = {0.f, 0.f, 0.f, 0.f, 0.f, 0.f, 0.f, 0.f};
        #pragma unroll
        for (int ks = 0; ks < 9; ++ks) {
            const v2f a = ld2(&sP[(wrow + lm) * 36 + ks * 4 + 2 * half]);
            const v2f b = ld2(&sYpk[((nt * 9 + ks) * 32 + lane) * 2]);
            acc = wmma_f32(a, b, acc);
        }
        #pragma unroll
        for (int j = 0; j < 8; ++j)
            sFG[(wrow + j + 8 * half) * 148 + nt * 16 + lm] = acc[j];
    }
    // result init: nu = 0 term (prod = feat)
    for (int i = tid; i < 2304; i += 128) {
        const int c = i / 36, k = i % 36;
        const int l = (int)sqrtf((float)k + 0.5f);   // floor(sqrt(k))
        sR[i] = sW[c * 6 + l] * sP[i];
    }
    __syncthreads();

    // ---- correlation loop nu = 1, 2 ---------------------------------------
    for (int nu = 1; nu <= 2; ++nu) {
        // stage B: prod_grid = to_grid(prod) * feat_grid
        for (int nt = 0; nt < 9; ++nt) {
            v8f acc = {0.f, 0.f, 0.f, 0.f, 0.f, 0.f, 0.f, 0.f};
            #pragma unroll
            for (int ks = 0; ks < 9; ++ks) {
                const v2f a = ld2(&sP[(wrow + lm) * 36 + ks * 4 + 2 * half]);
                const v2f b = ld2(&sYpk[((nt * 9 + ks) * 32 + lane) * 2]);
                acc = wmma_f32(a, b, acc);
            }
            #pragma unroll
            for (int j = 0; j < 8; ++j) {
                const int idx = (wrow + j + 8 * half) * 148 + nt * 16 + lm;
                sPG[idx] = acc[j] * sFG[idx];
            }
        }
        __syncthreads();

        // stage C: prod = from_grid(prod_grid)   (K = 144, N padded to 48)
        for (int nt = 0; nt < 3; ++nt) {
            v8f acc = {0.f, 0.f, 0.f, 0.f, 0.f, 0.f, 0.f, 0.f};
            #pragma unroll
            for (int ks = 0; ks < 36; ++ks) {
                const v2f a = ld2(&sPG[(wrow + lm) * 148 + ks * 4 + 2 * half]);
                const v2f b = ld2(&sYwpk[((nt * 36 + ks) * 32 + lane) * 2]);
                acc = wmma_f32(a, b, acc);
            }
            const int col = nt * 16 + lm;
            #pragma unroll
            for (int j = 0; j < 8; ++j) {
                if (col < 36)
                    sP[(wrow + j + 8 * half) * 36 + col] = acc[j];
            }
        }
        __syncthreads();

        // stage D: result += wn[nu] * prod
        for (int i = tid; i < 2304; i += 128) {
            const int c = i / 36, k = i % 36;
            const int l = (int)sqrtf((float)k + 0.5f);
            sR[i] += sW[(nu * 64 + c) * 6 + l] * sP[i];
        }
        __syncthreads();
    }

    // ---- writeback ---------------------------------------------------------
    for (int i = tid; i < 2304; i += 128) out[fbase + i] = sR[i];
}

extern "C" void kernel_launch(void* const* d_in, const int* in_sizes, int n_in,
                              void* d_out, int out_size, void* d_ws, size_t ws_size,
                              hipStream_t stream) {
    const float* atom_feat = (const float*)d_in[0];   // [N][128][36]
    const float* atom_type = (const float*)d_in[1];   // [N][10]
    const float* weights   = (const float*)d_in[2];   // [3][10][128][6]
    float* out = (float*)d_out;

    float* Ypk  = (float*)d_ws;       // 5184 floats
    float* Ywpk = Ypk + 5184;         // 6912 floats

    const int natoms = in_sizes[0] / (128 * 36);

    gaunt_setup<<<1, 256, 0, stream>>>(Ypk, Ywpk);

    const size_t smem = 36816 * sizeof(float);  // 147264 B dynamic LDS
    (void)hipFuncSetAttribute((const void*)gaunt_main,
                              hipFuncAttributeMaxDynamicSharedMemorySize,
                              (int)smem);
    gaunt_main<<<dim3(natoms * 2), dim3(128), smem, stream>>>(
        atom_feat, atom_type, weights, Ypk, Ywpk, out);
    (void)n_in; (void)out_size; (void)ws_size;
}